// ReEig_57415122813336
// MI455X (gfx1250) — compile-verified
//
#include <hip/hip_runtime.h>

// ReEig via Newton-Schulz matrix-sign iteration (GEMM-only formulation):
//   f(P) = 0.5 * (P + eps*I + |P - eps*I|),  |M| = M * sign(M)
//   sign via NS:  Y0 = M/||M||_F,  Y <- 0.5 * Y * (3I - Y*Y)
// All GEMMs are 64x64 fp32 done with V_WMMA_F32_16X16X4_F32 tiles.
// Every operand matrix in the chain is SYMMETRIC, so the B operand is
// fetched transposed (row-contiguous) -> both WMMA operands are single
// 64-bit LDS loads into even-aligned VGPR pairs. The (3I - T) transform
// is applied once on the producer store, not per consumer load.

typedef float v2f __attribute__((ext_vector_type(2)));
typedef float v8f __attribute__((ext_vector_type(8)));

#define MATN   64
#define LSTR   65        // LDS row stride (odd -> bank-conflict-free)
#define NITER  28        // Newton-Schulz iterations (spectral gap ~8e-5 needs ~25)
#define EPS_R  1e-4f

// One 16x16 C tile of C = A * B, A and B 64x64 SYMMETRIC in LDS (stride LSTR).
// B is read transposed: B[k][n] == B[n][k] (contiguous in k).
// ADIAG: A := A - eps*I
template <int ADIAG>
__device__ __forceinline__ v8f tile_mm64(const float* __restrict__ Ab,
                                         const float* __restrict__ Bb,
                                         int ti, int tj, int lane) {
    const int rc   = lane & 15;          // A row-in-tile / B col-in-tile
    const int koff = (lane >> 4) << 1;   // 0 or 2
    const int arow = ti * 16 + rc;
    const int bcol = tj * 16 + rc;
    v8f acc = {};
#pragma unroll
    for (int k = 0; k < MATN; k += 4) {
        const int k0 = k + koff;
        float a0 = Ab[arow * LSTR + k0];
        float a1 = Ab[arow * LSTR + k0 + 1];
        if (ADIAG) {
            if (arow == k0)     a0 -= EPS_R;
            if (arow == k0 + 1) a1 -= EPS_R;
        }
        // symmetric fetch: B[k0][bcol] = B[bcol][k0]  (contiguous pair)
        const float b0 = Bb[bcol * LSTR + k0];
        const float b1 = Bb[bcol * LSTR + k0 + 1];
        v2f av; av.x = a0; av.y = a1;
        v2f bv; bv.x = b0; bv.y = b1;
        acc = __builtin_amdgcn_wmma_f32_16x16x4_f32(
            /*neg_a=*/false, av, /*neg_b=*/false, bv,
            /*c_mod=*/(short)0, acc, /*reuse_a=*/false, /*reuse_b=*/false);
    }
    return acc;
}

// Scatter a 16x16 f32 C tile (ISA C/D layout) back to LDS.
// SMODE==1: store 3*(r==c) - val   (produces S = 3I - T in one pass)
template <int SMODE>
__device__ __forceinline__ void store_tile(float* __restrict__ dst, v8f c,
                                           int ti, int tj, int lane) {
    const int col   = tj * 16 + (lane & 15);
    const int rbase = ti * 16 + ((lane >> 4) << 3);
#pragma unroll
    for (int v = 0; v < 8; ++v) {
        const int r = rbase + v;
        float val = c[v];
        if (SMODE) val = ((r == col) ? 3.0f : 0.0f) - val;
        dst[r * LSTR + col] = val;
    }
}

__launch_bounds__(256, 2)
__global__ void reeig_ns_wmma_kernel(const float* __restrict__ P,
                                     float* __restrict__ out) {
    __shared__ float smP[MATN * LSTR];   // original P
    __shared__ float smY[MATN * LSTR];   // sign iterate
    __shared__ float smT[MATN * LSTR];   // S = 3I - Y*Y temp
    __shared__ float red[256];

    const int tid  = threadIdx.x;
    const int lane = tid & 31;
    const int wave = tid >> 5;

    const float* __restrict__ Pg = P   + (size_t)blockIdx.x * (MATN * MATN);
    float*       __restrict__ Og = out + (size_t)blockIdx.x * (MATN * MATN);

    // ---- Load P, build M = P - eps*I, accumulate ||M||_F^2 partials ----
    float ss = 0.0f;
#pragma unroll
    for (int i = 0; i < 16; ++i) {
        const int idx = tid * 16 + i;
        const int r = idx >> 6, c = idx & 63;
        const float v = Pg[idx];
        smP[r * LSTR + c] = v;
        const float m = v - ((r == c) ? EPS_R : 0.0f);
        smY[r * LSTR + c] = m;
        ss += m * m;
    }
    red[tid] = ss;
    __syncthreads();
    for (int off = 128; off > 0; off >>= 1) {
        if (tid < off) red[tid] += red[tid + off];
        __syncthreads();
    }
    const float inv_s = 1.0f / sqrtf(red[0] + 1e-30f);   // 1/||M||_F  (=> ||Y0||_2 <= 1)
    __syncthreads();
#pragma unroll
    for (int i = 0; i < 16; ++i) {
        const int idx = tid * 16 + i;
        const int r = idx >> 6, c = idx & 63;
        smY[r * LSTR + c] *= inv_s;
    }
    __syncthreads();

    // ---- 8 waves x 2 tiles = the 4x4 grid of 16x16 tiles ----
    const int t0 = wave * 2, t1 = t0 + 1;
    const int ti0 = t0 >> 2, tj0 = t0 & 3;   // t0,t1 share ti -> shared A fetches
    const int ti1 = t1 >> 2, tj1 = t1 & 3;

    // ---- Newton-Schulz: Y <- 0.5 * Y * (3I - Y*Y) ----
    for (int it = 0; it < NITER; ++it) {
        v8f c0 = tile_mm64<0>(smY, smY, ti0, tj0, lane);   // T = Y*Y
        v8f c1 = tile_mm64<0>(smY, smY, ti1, tj1, lane);
        store_tile<1>(smT, c0, ti0, tj0, lane);            // smT := 3I - T
        store_tile<1>(smT, c1, ti1, tj1, lane);
        __syncthreads();
        v8f d0 = tile_mm64<0>(smY, smT, ti0, tj0, lane);   // Y * S
        v8f d1 = tile_mm64<0>(smY, smT, ti1, tj1, lane);
        __syncthreads();
#pragma unroll
        for (int v = 0; v < 8; ++v) { d0[v] *= 0.5f; d1[v] *= 0.5f; }
        store_tile<0>(smY, d0, ti0, tj0, lane);
        store_tile<0>(smY, d1, ti1, tj1, lane);
        __syncthreads();
    }

    // ---- R = (P - eps*I) * sign ;  out = 0.5 * (P + eps*I + R) ----
    v8f r0 = tile_mm64<1>(smP, smY, ti0, tj0, lane);
    v8f r1 = tile_mm64<1>(smP, smY, ti1, tj1, lane);

    {
        const int col   = tj0 * 16 + (lane & 15);
        const int rbase = ti0 * 16 + ((lane >> 4) << 3);
#pragma unroll
        for (int v = 0; v < 8; ++v) {
            const int r = rbase + v;
            const float pv = smP[r * LSTR + col] + ((r == col) ? EPS_R : 0.0f);
            Og[r * MATN + col] = 0.5f * (pv + r0[v]);
        }
    }
    {
        const int col   = tj1 * 16 + (lane & 15);
        const int rbase = ti1 * 16 + ((lane >> 4) << 3);
#pragma unroll
        for (int v = 0; v < 8; ++v) {
            const int r = rbase + v;
            const float pv = smP[r * LSTR + col] + ((r == col) ? EPS_R : 0.0f);
            Og[r * MATN + col] = 0.5f * (pv + r1[v]);
        }
    }
}

extern "C" void kernel_launch(void* const* d_in, const int* in_sizes, int n_in,
                              void* d_out, int out_size, void* d_ws, size_t ws_size,
                              hipStream_t stream) {
    const float* P = (const float*)d_in[0];
    float* out     = (float*)d_out;
    const int B    = in_sizes[0] / (MATN * MATN);   // 8192
    reeig_ns_wmma_kernel<<<dim3(B), dim3(256), 0, stream>>>(P, out);
}